// ours_temp_91173565760144
// MI455X (gfx1250) — compile-verified
//
#include <hip/hip_runtime.h>
#include <math.h>

#define NW 20000
#define NT 50
#define ND 2000
#define NTOT 22050
#define EDGES 150000
#define MT_W 1250            // 20000/16
#define MT_T 4               // ceil(50/16)
#define MT_D 125             // 2000/16
#define MT_TOT (MT_W + MT_T + MT_D)

typedef __attribute__((ext_vector_type(2))) float v2f;
typedef __attribute__((ext_vector_type(8))) float v8f;

__device__ __forceinline__ v8f wmma4(v2f a, v2f b, v8f c) {
  // D = A(16x4,f32) * B(4x16,f32) + C(16x16,f32)
  return __builtin_amdgcn_wmma_f32_16x16x4_f32(false, a, false, b, (short)0, c, false, false);
}

__global__ void zero_kernel(float* __restrict__ p, int n) {
  int i = blockIdx.x * blockDim.x + threadIdx.x;
  if (i < n) p[i] = 0.f;
}

// One wave computes one 16x16 tile of out = X[Nt,256] @ W[256,128] + b.
// blockDim=256 (8 waves, wave w handles N-tile w). blockIdx.y = 0/1/2 -> k/q/v.
__global__ void proj_kernel(const float* __restrict__ xw, const float* __restrict__ xt,
                            const float* __restrict__ xd,
                            const float* __restrict__ Wk, const float* __restrict__ bk,
                            const float* __restrict__ Wq, const float* __restrict__ bq,
                            const float* __restrict__ Wv, const float* __restrict__ bv,
                            float* __restrict__ ok, float* __restrict__ oq,
                            float* __restrict__ ov)
{
  int mt = blockIdx.x;
  int nt, mrow, nrows, obase;
  const float* X;
  if (mt < MT_W)            { nt = 0; X = xw; nrows = NW; mrow = mt * 16;              obase = 0;       }
  else if (mt < MT_W + MT_T){ nt = 1; X = xt; nrows = NT; mrow = (mt - MT_W) * 16;     obase = NW;      }
  else                      { nt = 2; X = xd; nrows = ND; mrow = (mt - MT_W - MT_T) * 16; obase = NW + NT; }

  int p = blockIdx.y;
  const float* W = (p == 0 ? Wk : p == 1 ? Wq : Wv) + nt * 256 * 128;
  const float* B = (p == 0 ? bk : p == 1 ? bq : bv) + nt * 128;
  float* O       = (p == 0 ? ok : p == 1 ? oq : ov);

  int wv   = threadIdx.x >> 5;   // N tile 0..7
  int lane = threadIdx.x & 31;
  int half = lane >> 4;          // 0: K0/K1, 1: K2/K3
  int lr   = lane & 15;
  int ncol = wv * 16 + lr;

  int arow = mrow + lr;
  if (arow >= nrows) arow = nrows - 1;     // clamp reads; OOB rows never stored
  const float* Ar = X + (size_t)arow * 256;

  v8f c = {0.f, 0.f, 0.f, 0.f, 0.f, 0.f, 0.f, 0.f};
  for (int k0 = 0; k0 < 256; k0 += 4) {
    int kk = k0 + half * 2;
    v2f a; a.x = Ar[kk];                 a.y = Ar[kk + 1];
    v2f b; b.x = W[kk * 128 + ncol];     b.y = W[(kk + 1) * 128 + ncol];
    c = wmma4(a, b, c);
  }
  float bias = B[ncol];
#pragma unroll
  for (int i = 0; i < 8; i++) {
    int row = mrow + i + half * 8;       // C: vgpr i -> M=i (lanes 0-15) / M=i+8 (16-31)
    if (row < nrows) O[(size_t)(obase + row) * 128 + ncol] = c[i] + bias;
  }
}

// Pass 1 per etype: att dot, exp, atomic denominator. One thread per (edge, head).
__global__ void edge_att_kernel(const int* __restrict__ src, const int* __restrict__ dst,
                                const float* __restrict__ kbuf, const float* __restrict__ qbuf,
                                const float* __restrict__ pri_e,   // pri + e*8
                                int sbase, int dbase,
                                float* __restrict__ ex, float* __restrict__ den)
{
  int idx = blockIdx.x * blockDim.x + threadIdx.x;
  if (idx >= EDGES * 8) return;
  int h = idx & 7;
  int i = idx >> 3;
  int s = src[i], d = dst[i];
  const float4* q4 = (const float4*)(qbuf + (size_t)(dbase + d) * 128 + h * 16);
  const float4* k4 = (const float4*)(kbuf + (size_t)(sbase + s) * 128 + h * 16);
  float att = 0.f;
#pragma unroll
  for (int t = 0; t < 4; t++) {
    float4 qv = q4[t], kv = k4[t];
    att += qv.x * kv.x + qv.y * kv.y + qv.z * kv.z + qv.w * kv.w;
  }
  att = att * pri_e[h] * 0.25f;          // / sqrt(DK=16)
  float e = expf(att);                   // softmax is shift-invariant; |att| small
  ex[idx] = e;
  atomicAdd(&den[d * 8 + h], e);
}

// Pass 2 per etype: val = v @ msg[e,h] via WMMA, scale by a=ex/den, atomic into t_acc.
// blockDim=256: wave = head, each block covers 16 edges.
__global__ void edge_val_kernel(const int* __restrict__ src, const int* __restrict__ dst,
                                const float* __restrict__ vbuf,
                                const float* __restrict__ msg_e,   // msg + e*2048
                                const float* __restrict__ ex, const float* __restrict__ den,
                                int sbase, int dbase, float* __restrict__ tacc)
{
  int h    = threadIdx.x >> 5;
  int lane = threadIdx.x & 31;
  int half = lane >> 4;
  int lr   = lane & 15;
  int e0   = blockIdx.x * 16;            // EDGES % 16 == 0

  int s = src[e0 + lr];
  const float* vp = vbuf + (size_t)(sbase + s) * 128 + h * 16;   // A row: v[edge,h,:]

  v8f c = {0.f, 0.f, 0.f, 0.f, 0.f, 0.f, 0.f, 0.f};
#pragma unroll
  for (int k0 = 0; k0 < 16; k0 += 4) {
    int kk = k0 + half * 2;
    v2f a; a.x = vp[kk];                          a.y = vp[kk + 1];
    v2f b; b.x = msg_e[h * 256 + kk * 16 + lr];   b.y = msg_e[h * 256 + (kk + 1) * 16 + lr];
    c = wmma4(a, b, c);
  }
#pragma unroll
  for (int i = 0; i < 8; i++) {
    int ge = e0 + i + half * 8;
    int d  = dst[ge];
    float a_eh = ex[ge * 8 + h] / den[d * 8 + h];
    atomicAdd(&tacc[(size_t)(dbase + d) * 128 + h * 16 + lr], a_eh * c[i]);
  }
}

// One wave32 per node: mean-divide + relu + LayerNorm (128 dims -> 4 per lane).
__global__ void finalize_kernel(const float* __restrict__ tacc,
                                const float* __restrict__ gamma,
                                const float* __restrict__ beta,
                                float* __restrict__ out)
{
  int node = blockIdx.x * 8 + (threadIdx.x >> 5);
  if (node >= NTOT) return;
  int lane = threadIdx.x & 31;
  int nt = node < NW ? 0 : (node < NW + NT ? 1 : 2);
  float inv_cnt = (nt == 2) ? 0.5f : (1.f / 3.f);  // DST_COUNT = 3,3,2

  float vals[4];
  float sum = 0.f;
#pragma unroll
  for (int j = 0; j < 4; j++) {
    float x = tacc[(size_t)node * 128 + j * 32 + lane] * inv_cnt;
    x = x > 0.f ? x : 0.f;
    vals[j] = x;
    sum += x;
  }
#pragma unroll
  for (int off = 16; off > 0; off >>= 1) sum += __shfl_xor(sum, off, 32);
  float mu = sum * (1.f / 128.f);
  float var = 0.f;
#pragma unroll
  for (int j = 0; j < 4; j++) { float dd = vals[j] - mu; var += dd * dd; }
#pragma unroll
  for (int off = 16; off > 0; off >>= 1) var += __shfl_xor(var, off, 32);
  var *= (1.f / 128.f);
  float inv = 1.f / sqrtf(var + 1e-5f);
#pragma unroll
  for (int j = 0; j < 4; j++) {
    int col = j * 32 + lane;
    out[(size_t)node * 128 + col] =
        (vals[j] - mu) * inv * gamma[nt * 128 + col] + beta[nt * 128 + col];
  }
}

extern "C" void kernel_launch(void* const* d_in, const int* in_sizes, int n_in,
                              void* d_out, int out_size, void* d_ws, size_t ws_size,
                              hipStream_t stream)
{
  (void)in_sizes; (void)n_in; (void)out_size; (void)ws_size;
  const float* xw   = (const float*)d_in[0];
  const float* xt   = (const float*)d_in[1];
  const float* xd   = (const float*)d_in[2];
  const float* Wk   = (const float*)d_in[3];
  const float* bk   = (const float*)d_in[4];
  const float* Wq   = (const float*)d_in[5];
  const float* bq   = (const float*)d_in[6];
  const float* Wv   = (const float*)d_in[7];
  const float* bv   = (const float*)d_in[8];
  const float* pri  = (const float*)d_in[9];
  const float* msg  = (const float*)d_in[10];
  const float* gam  = (const float*)d_in[11];
  const float* bet  = (const float*)d_in[12];
  const int*   esrc = (const int*)d_in[13];
  const int*   edst = (const int*)d_in[14];
  float* out = (float*)d_out;

  float* ws   = (float*)d_ws;
  float* kb   = ws;                                 // [22050,128]
  float* qb   = kb   + (size_t)NTOT * 128;          // [22050,128]
  float* vb   = qb   + (size_t)NTOT * 128;          // [22050,128]
  float* tacc = vb   + (size_t)NTOT * 128;          // [22050,128]
  float* den  = tacc + (size_t)NTOT * 128;          // [<=20000,8], reused per etype
  float* ex   = den  + (size_t)NW * 8;              // [150000,8], reused per etype

  { int n = NTOT * 128;
    zero_kernel<<<(n + 255) / 256, 256, 0, stream>>>(tacc, n); }

  proj_kernel<<<dim3(MT_TOT, 3, 1), 256, 0, stream>>>(xw, xt, xd, Wk, bk, Wq, bq, Wv, bv,
                                                      kb, qb, vb);

  const int srcT[8] = {0, 0, 1, 1, 0, 1, 2, 2};
  const int dstT[8] = {1, 2, 2, 1, 0, 0, 1, 0};
  const int base[3] = {0, NW, NW + NT};
  const int nn[3]   = {NW, NT, ND};

  for (int e = 0; e < 8; e++) {
    int st = srcT[e], dt = dstT[e];
    int nden = nn[dt] * 8;
    zero_kernel<<<(nden + 255) / 256, 256, 0, stream>>>(den, nden);
    const int* se = esrc + (size_t)e * EDGES;
    const int* de = edst + (size_t)e * EDGES;
    int natt = EDGES * 8;
    edge_att_kernel<<<(natt + 255) / 256, 256, 0, stream>>>(se, de, kb, qb, pri + e * 8,
                                                            base[st], base[dt], ex, den);
    edge_val_kernel<<<EDGES / 16, 256, 0, stream>>>(se, de, vb, msg + e * 2048, ex, den,
                                                    base[st], base[dt], tacc);
  }

  finalize_kernel<<<(NTOT + 7) / 8, 256, 0, stream>>>(tacc, gam, bet, out);
}